// MultiDilateWindowAttention_3865470566730
// MI455X (gfx1250) — compile-verified
//
#include <hip/hip_runtime.h>
#include <hip/hip_bf16.h>

// Problem constants (from reference setup_inputs)
#define BB 4
#define CC 192
#define HH 128
#define WWID 128
#define PP (HH * WWID)          // 16384 pixels per batch
#define ND 3                    // number of dilations (1,2,3)
#define C2 (CC / ND)            // 64 channels per dilation group
#define HD 32                   // head dim (C / NUM_HEADS = 192/6)
#define NH_PER_D (C2 / HD)      // 2 heads per dilation

typedef __attribute__((ext_vector_type(2))) float v2f;
typedef __attribute__((ext_vector_type(8))) float v8f;

__device__ __forceinline__ int reflect_idx(int i, int n) {
    i = (i < 0) ? -i : i;
    i = (i >= n) ? (2 * n - 2 - i) : i;
    return i;
}

// ---------------------------------------------------------------------------
// Kernel 1: depthwise 3x3 positional conv (zero pad) + bias + residual
//   xp[b,c,p] = x[b,c,p] + sum_{kh,kw} pw[c,kh,kw]*x[b,c,zpad(h+kh-1,w+kw-1)] + pb[c]
// grid: (PP/256, CC, BB), block 256; lane-per-pixel -> coalesced
// ---------------------------------------------------------------------------
__global__ void __launch_bounds__(256)
k_pos_embed(const float* __restrict__ x, const float* __restrict__ pw,
            const float* __restrict__ pb, float* __restrict__ xp)
{
    const int p = blockIdx.x * 256 + threadIdx.x;
    const int c = blockIdx.y;
    const int b = blockIdx.z;
    const int h = p >> 7;
    const int w = p & 127;
    const float* xc = x + ((size_t)(b * CC + c)) * PP;
    const float* wc = pw + c * 9;
    float acc = 0.f;
#pragma unroll
    for (int kh = 0; kh < 3; ++kh) {
        const int hy = h + kh - 1;
        if (hy < 0 || hy >= HH) continue;
#pragma unroll
        for (int kw = 0; kw < 3; ++kw) {
            const int wx = w + kw - 1;
            if (wx < 0 || wx >= WWID) continue;
            acc = fmaf(wc[kh * 3 + kw], xc[hy * WWID + wx], acc);
        }
    }
    xp[((size_t)(b * CC + c)) * PP + p] = xc[p] + acc + pb[c];
}

// ---------------------------------------------------------------------------
// Kernel 2: f32 WMMA GEMM  Y[b] (M x PP) = W (M x K, row major) @ X[b] (K x PP)
// Per wave: 64M x 64N tile = 4x4 grid of 16x16 accumulators,
//   K advanced 4 per step via V_WMMA_F32_16X16X4_F32 (16 WMMAs / step).
// 8 waves/block cover 512 N columns. grid: (PP/512, M/64, BB)
// A lane layout (ISA 16x4 f32): M = lane%16, K = (lane/16)*2 + vgpr  -> v2f load
// B lane layout (mirror):       N = lane%16, K = (lane/16)*2 + vgpr
// C/D layout: M = vgpr + 8*(lane/16), N = lane%16
// ---------------------------------------------------------------------------
__global__ void __launch_bounds__(256)
k_wmma_gemm(const float* __restrict__ Wm,   // [M, K] row-major weights
            const float* __restrict__ Xm,   // batched [K, PP]
            float* __restrict__ Ym,         // batched [M, PP]
            int K, long xStride, long yStride)
{
    const int lane = threadIdx.x & 31;
    const int wave = threadIdx.x >> 5;           // 0..7
    const int lmod = lane & 15;
    const int lhi  = lane >> 4;                  // 0 or 1
    const int mBase = blockIdx.y << 6;           // 64 rows per block
    const int nBase = (blockIdx.x << 9) + (wave << 6);  // 64 cols per wave
    const float* Xb = Xm + (size_t)blockIdx.z * xStride;
    float*       Yb = Ym + (size_t)blockIdx.z * yStride;

    v8f acc[4][4];
#pragma unroll
    for (int i = 0; i < 4; ++i)
#pragma unroll
        for (int j = 0; j < 4; ++j) acc[i][j] = (v8f){};

    const float* aPtr = Wm + (size_t)(mBase + lmod) * K + (lhi << 1);
    const float* bPtr = Xb + (size_t)(lhi << 1) * PP + nBase + lmod;

    for (int k0 = 0; k0 < K; k0 += 4) {
        v2f a[4];
#pragma unroll
        for (int i = 0; i < 4; ++i)
            a[i] = *(const v2f*)(aPtr + (size_t)(i << 4) * K);  // rows mBase+i*16+lmod
        aPtr += 4;

        v2f bmat[4];
#pragma unroll
        for (int j = 0; j < 4; ++j) {
            bmat[j].x = bPtr[(j << 4)];
            bmat[j].y = bPtr[(j << 4) + PP];
        }
        bPtr += (size_t)4 * PP;

#pragma unroll
        for (int i = 0; i < 4; ++i)
#pragma unroll
            for (int j = 0; j < 4; ++j)
                acc[i][j] = __builtin_amdgcn_wmma_f32_16x16x4_f32(
                    false, a[i], false, bmat[j], (short)0, acc[i][j], false, false);
    }

#pragma unroll
    for (int i = 0; i < 4; ++i) {
#pragma unroll
        for (int r = 0; r < 8; ++r) {
            const int m = mBase + (i << 4) + (lhi << 3) + r;
            float* yr = Yb + (size_t)m * PP + nBase + lmod;
#pragma unroll
            for (int j = 0; j < 4; ++j)
                yr[j << 4] = acc[i][j][r];
        }
    }
}

// ---------------------------------------------------------------------------
// Kernel 3: dilated 3x3 window attention with reflection padding.
// One thread = one (b, dilation, head, pixel). head_dim=32, 9 taps.
// qkv layout (from GEMM): [B, 3, ND, C2, PP]; xo: [B, ND*C2, PP] dilation-major.
// grid: (PP/256, ND*NH_PER_D, BB)
// ---------------------------------------------------------------------------
__global__ void __launch_bounds__(256)
k_dilate_attn(const float* __restrict__ qkv, float* __restrict__ xo)
{
    const int p = blockIdx.x * 256 + threadIdx.x;
    const int hidx = blockIdx.y;           // 0..5
    const int d = hidx >> 1;               // dilation index
    const int head = hidx & 1;
    const int b = blockIdx.z;
    const int dil = d + 1;
    const int h = p >> 7;
    const int w = p & 127;

    int nidx[9];
#pragma unroll
    for (int kh = 0; kh < 3; ++kh) {
        const int rh = reflect_idx(h + (kh - 1) * dil, HH);
#pragma unroll
        for (int kw = 0; kw < 3; ++kw) {
            const int rw = reflect_idx(w + (kw - 1) * dil, WWID);
            nidx[kh * 3 + kw] = rh * WWID + rw;
        }
    }

    const size_t base = (size_t)b * 3 * CC * PP;
    const size_t chan = (size_t)(d * C2 + head * HD);
    const float* qp = qkv + base + (0 * CC + chan) * PP + p;
    const float* kp = qkv + base + (1 * CC + chan) * PP;
    const float* vp = qkv + base + (2 * CC + chan) * PP;

    float s[9];
#pragma unroll
    for (int j = 0; j < 9; ++j) s[j] = 0.f;
    for (int c = 0; c < HD; ++c) {
        const float qv = qp[(size_t)c * PP];
        const float* kc = kp + (size_t)c * PP;
#pragma unroll
        for (int j = 0; j < 9; ++j) s[j] = fmaf(qv, kc[nidx[j]], s[j]);
    }

    const float scale = 0.17677669529663687f;  // 32^-0.5
    float mx = s[0];
#pragma unroll
    for (int j = 1; j < 9; ++j) mx = fmaxf(mx, s[j]);
    float sum = 0.f;
#pragma unroll
    for (int j = 0; j < 9; ++j) {
        s[j] = __expf((s[j] - mx) * scale);
        sum += s[j];
    }
    const float inv = 1.f / sum;
#pragma unroll
    for (int j = 0; j < 9; ++j) s[j] *= inv;

    float* op = xo + ((size_t)b * CC + chan) * PP + p;
    for (int c = 0; c < HD; ++c) {
        const float* vc = vp + (size_t)c * PP;
        float acc = 0.f;
#pragma unroll
        for (int j = 0; j < 9; ++j) acc = fmaf(s[j], vc[nidx[j]], acc);
        op[(size_t)c * PP] = acc;
    }
}

// ---------------------------------------------------------------------------
extern "C" void kernel_launch(void* const* d_in, const int* in_sizes, int n_in,
                              void* d_out, int out_size, void* d_ws, size_t ws_size,
                              hipStream_t stream) {
    const float* x      = (const float*)d_in[0];   // [B, C, H, W]
    const float* pos_w  = (const float*)d_in[1];   // [C, 1, 3, 3]
    const float* pos_b  = (const float*)d_in[2];   // [C]
    const float* qkv_w  = (const float*)d_in[3];   // [3C, C]
    const float* proj_w = (const float*)d_in[4];   // [C, C]
    float* out = (float*)d_out;                    // [B, C, H, W]

    float* xp   = (float*)d_ws;                    // [B, C, P]     50.3 MB
    float* qkv  = xp  + (size_t)BB * CC * PP;      // [B, 3C, P]   150.9 MB
    float* xo   = qkv + (size_t)BB * 3 * CC * PP;  // [B, C, P]     50.3 MB

    dim3 blk(256);

    // 1) positional depthwise conv + residual + bias
    k_pos_embed<<<dim3(PP / 256, CC, BB), blk, 0, stream>>>(x, pos_w, pos_b, xp);

    // 2) qkv 1x1 conv: [576 x 192] @ [192 x 16384] per batch  (WMMA f32)
    k_wmma_gemm<<<dim3(PP / 512, (3 * CC) / 64, BB), blk, 0, stream>>>(
        qkv_w, xp, qkv, CC, (long)CC * PP, (long)3 * CC * PP);

    // 3) multi-dilation window attention
    k_dilate_attn<<<dim3(PP / 256, ND * NH_PER_D, BB), blk, 0, stream>>>(qkv, xo);

    // 4) proj 1x1 conv: [192 x 192] @ [192 x 16384] per batch (WMMA f32)
    k_wmma_gemm<<<dim3(PP / 512, CC / 64, BB), blk, 0, stream>>>(
        proj_w, xo, out, CC, (long)CC * PP, (long)CC * PP);
}